// EditDistance_26259430047955
// MI455X (gfx1250) — compile-verified
//
#include <hip/hip_runtime.h>
#include <stdint.h>

// Batched Levenshtein edit distance, one wave32 per problem (column n).
// DP row (R+1=1025 entries) lives in registers: 33 floats per lane, blocked.
// Deletion propagation = in-lane running min + wave32 exclusive prefix-min
// (positionally normalized). Hyp column staged into LDS via CDNA5 TDM
// (tensor_load_to_lds), overlapped with ref token loads.

#define EOS_TOK 0
#define LANES 32
#define CHUNK 33   // 32*33 = 1056 >= R+1 = 1025
#define HMAX 1024

typedef uint32_t u32x4 __attribute__((ext_vector_type(4)));
typedef int32_t  i32x8 __attribute__((ext_vector_type(8)));

__device__ __forceinline__ int imin(int a, int b) { return a < b ? a : b; }

__global__ __launch_bounds__(LANES) void EditDistance_cdna5_kernel(
    const int* __restrict__ ref, const int* __restrict__ hyp,
    const float* __restrict__ costs, float* __restrict__ out,
    int R, int H, int N) {
  __shared__ int s_hyp[HMAX];

  const int lane = threadIdx.x;
  const int n = blockIdx.x;           // problem index (wave-uniform, scalar)
  const int Hc = imin(H, HMAX);

  // ---------------- TDM: gather hyp column n (stride N words) into LDS ----
  {
    uint64_t gaddr = (uint64_t)(uintptr_t)hyp + 4ull * (uint32_t)n;
    uint32_t ldsoff = (uint32_t)(uintptr_t)(&s_hyp[0]);   // low 32 bits = LDS offset

    u32x4 g0;
    g0[0] = 1u;                                            // count=1, user descriptor
    g0[1] = ldsoff;                                        // lds_addr
    g0[2] = (uint32_t)gaddr;                               // global_addr[31:0]
    g0[3] = (uint32_t)((gaddr >> 32) & 0x1FFFFFFull)       // global_addr[56:32]
          | (2u << 30);                                    // type = 2 ("image")

    i32x8 g1;
    g1[0] = (int)(2u << 16);                               // data_size = 4 bytes
    g1[1] = (int)(1u << 16);                               // tensor_dim0 = 1
    g1[2] = (int)((uint32_t)Hc << 16);                     // tensor_dim1 = Hc
    g1[3] = (int)(1u << 16);                               // tile_dim0 = 1
    g1[4] = (int)(uint32_t)Hc;                             // tile_dim1 = Hc rows, tile_dim2=0
    g1[5] = (int)N;                                        // tensor_dim0_stride = N
    g1[6] = 0;                                             // stride hi / dim1_stride lo
    g1[7] = 0;                                             // dim1_stride hi (unused, 2D)

    asm volatile("tensor_load_to_lds %0, %1" :: "s"(g0), "s"(g1) : "memory");
  }

  // ---------------- Costs (wave-uniform scalar loads) ---------------------
  const float ins_c = costs[0];
  const float del_c = costs[1];
  const float sub_c = costs[2];

  // ---------------- Init DP row + load ref tokens (overlaps TDM) ----------
  const int base_i = lane * CHUNK;    // this lane owns cells i in [base_i, base_i+32]
  float rowv[CHUNK];
  int tok[CHUNK];                     // token for cell i is ref[i-1][n]
  int firstEos = R;

#pragma unroll
  for (int k = 0; k < CHUNK; ++k) {
    const int i = base_i + k;
    rowv[k] = del_c * (float)i;       // row0 = deletion baseline
    const int ti = i - 1;
    int tv = -1;                      // sentinel: never matches a live hyp token
    if (ti >= 0 && ti < R) tv = ref[(size_t)ti * (size_t)N + (size_t)n];
    tok[k] = tv;
    if (tv == EOS_TOK && ti >= 0 && ti < R) firstEos = imin(firstEos, ti);
  }
  // wave min-reduce -> ref_len on all lanes
#pragma unroll
  for (int off = 16; off > 0; off >>= 1)
    firstEos = imin(firstEos, __shfl_xor(firstEos, off));
  const int ref_len = firstEos;

  // ---------------- Wait for TDM, then run the DP -------------------------
  asm volatile("s_wait_tensorcnt 0x0" ::: "memory");

  for (int t = 1; t <= Hc; ++t) {
    const int h = s_hyp[t - 1];       // LDS broadcast, wave-uniform
    if (h == EOS_TOK) break;          // steps with (t-1) >= hyp_len are no-ops

    // old value at i = base_i - 1 (previous lane's last cell)
    const float carry_old = __shfl_up(rowv[CHUNK - 1], 1u);

    // in-lane: insertion/substitution + local deletion chain
    float prevold = carry_old;
    float g = 0.0f;
#pragma unroll
    for (int k = 0; k < CHUNK; ++k) {
      const float oldv = rowv[k];
      float s = oldv + ins_c;                       // insertion
      if (!(k == 0 && lane == 0) || k != 0) {       // cell i==0 has no substitution
        const float sv = prevold + (tok[k] != h ? sub_c : 0.0f);
        if (k != 0 || lane != 0) s = fminf(s, sv);
      }
      g = (k == 0) ? s : fminf(s, g + del_c);       // local running deletion min
      rowv[k] = g;
      prevold = oldv;
    }

    // cross-lane deletion carry: exclusive prefix-min of P_l = T_l - del*i_last
    float p = g - del_c * (float)(base_i + CHUNK - 1);
#pragma unroll
    for (int off = 1; off < LANES; off <<= 1) {
      const float o = __shfl_up(p, (unsigned)off);
      if (lane >= off) p = fminf(p, o);
    }
    const float e = __shfl_up(p, 1u);               // exclusive: min over lanes < me
    if (lane > 0) {
#pragma unroll
      for (int k = 0; k < CHUNK; ++k)
        rowv[k] = fminf(rowv[k], e + del_c * (float)(base_i + k));
    }
  }

  // ---------------- Emit row[ref_len] -------------------------------------
  float result = 0.0f;
#pragma unroll
  for (int k = 0; k < CHUNK; ++k)
    if (base_i + k == ref_len) result = rowv[k];
  if (lane == ref_len / CHUNK) out[n] = result;
}

extern "C" void kernel_launch(void* const* d_in, const int* in_sizes, int n_in,
                              void* d_out, int out_size, void* d_ws, size_t ws_size,
                              hipStream_t stream) {
  const int* ref = (const int*)d_in[0];
  const int* hyp = (const int*)d_in[1];
  const float* costs = (const float*)d_in[2];
  float* out = (float*)d_out;

  const int N = out_size;                 // 2048 problems
  const int R = in_sizes[0] / N;          // 1024
  const int H = in_sizes[1] / N;          // 1024

  EditDistance_cdna5_kernel<<<N, LANES, 0, stream>>>(ref, hyp, costs, out, R, H, N);
}